// FNO_44839458570645
// MI455X (gfx1250) — compile-verified
//
#include <hip/hip_runtime.h>

typedef float v2f __attribute__((ext_vector_type(2)));
typedef float v8f __attribute__((ext_vector_type(8)));

static __device__ inline v8f wmma4(v2f a, v2f b, v8f c) {
  // D = A(16x4) * B(4x16) + C(16x16), all f32, wave32
  return __builtin_amdgcn_wmma_f32_16x16x4_f32(false, a, false, b, (short)0, c,
                                               false, false);
}
static __device__ inline v8f vzero8() {
  v8f z = {0.f, 0.f, 0.f, 0.f, 0.f, 0.f, 0.f, 0.f};
  return z;
}

#define PART_STRIDE 4194304u  // 2048 modes * 128 i * 16 b (floats)
#define TWO_PI_OVER_N 0.04908738521234052f

// ---------------- Kernel 1a: row real-DFT of z: R[b][x][ky], ky in [0,32) ---
__global__ __launch_bounds__(256) void fno_dft_rows(const float* __restrict__ z,
                                                    float* __restrict__ Rre,
                                                    float* __restrict__ Rim) {
  int idx = blockIdx.x * 256 + threadIdx.x;  // 65536 = 16*128*32
  int ky = idx & 31;
  int x = (idx >> 5) & 127;
  int b = idx >> 12;
  const float* zp = z + ((size_t)b * 128 + x) * 128;
  float sre = 0.f, sim = 0.f;
  for (int y = 0; y < 128; ++y) {
    float ang = -TWO_PI_OVER_N * (float)((ky * y) & 127);
    float s, c;
    __sincosf(ang, &s, &c);
    float v = zp[y];
    sre += v * c;
    sim += v * s;
  }
  Rre[idx] = sre;
  Rim[idx] = sim;
}

// ------------- Kernel 1b: column DFT at corner kx: Z[b][kxp][ky] ------------
__global__ __launch_bounds__(256) void fno_dft_cols(const float* __restrict__ Rre,
                                                    const float* __restrict__ Rim,
                                                    float* __restrict__ Zre,
                                                    float* __restrict__ Zim) {
  int idx = blockIdx.x * 256 + threadIdx.x;  // 32768 = 16*64*32
  int ky = idx & 31;
  int kxp = (idx >> 5) & 63;
  int b = idx >> 11;
  int kx = kxp + ((kxp >= 32) ? 64 : 0);  // kxp 32..63 -> kx 96..127
  float sre = 0.f, sim = 0.f;
  for (int x = 0; x < 128; ++x) {
    size_t ri = ((size_t)b * 128 + x) * 32 + ky;
    float re = Rre[ri], im = Rim[ri];
    float ang = -TWO_PI_OVER_N * (float)((kx * x) & 127);
    float s, c;
    __sincosf(ang, &s, &c);
    sre += re * c - im * s;
    sim += re * s + im * c;
  }
  Zre[idx] = sre;
  Zim[idx] = sim;
}

// ------------- Kernel 2: lift into spectrum  vf[part][mode][i][b] -----------
__global__ __launch_bounds__(256) void fno_lift(const float* __restrict__ Zre,
                                                const float* __restrict__ Zim,
                                                const float* __restrict__ lw,
                                                const float* __restrict__ lb,
                                                float* __restrict__ vf) {
  int idx = blockIdx.x * 256 + threadIdx.x;  // 8388608
  int b = idx & 15;
  int i = (idx >> 4) & 127;
  int mode = (idx >> 11) & 2047;
  int part = idx >> 22;
  float zv = (part == 0 ? Zre : Zim)[b * 2048 + mode];
  float val = lw[i] * zv;
  if (part == 0 && mode == 0) val += lb[i] * 16384.0f;  // DC of constant bias
  vf[idx] = val;
}

// -------- Kernel 3: per-mode complex channel mix, WMMA f32 16x16x4 ----------
// Grid 256 blocks: blockIdx.x = kxp*4 + myblock. Block 256 thr = 8 waves.
// Wave w handles o-tile [16w,16w+16). 8 my-modes per block for weight reuse.
template <bool FINAL>
__global__ __launch_bounds__(256) void fno_mix(const float* __restrict__ vin,
                                               float* __restrict__ vout,
                                               const float* __restrict__ wr1,
                                               const float* __restrict__ wi1,
                                               const float* __restrict__ wr2,
                                               const float* __restrict__ wi2) {
  extern __shared__ float lds[];  // 16 planes * 2048 floats = 128 KB
  const int bx = blockIdx.x;
  const int kxp = bx >> 2;        // 0..63
  const int my0 = (bx & 3) * 8;   // 0,8,16,24
  const int tid = threadIdx.x;

  // Stage activations asynchronously into LDS (no VGPR round-trip):
  // plane p = 2*j+part holds A[i(128)][b(16)] for mode my0+j.
  {
    const unsigned lds_base = (unsigned)(size_t)(&lds[0]);  // addr[31:0] = LDS offset
    for (int t = tid; t < 8192; t += 256) {  // 16B chunks
      int plane = t >> 9;
      int q = t & 511;
      int j = plane >> 1;
      int part = plane & 1;
      const float* src = vin + (size_t)part * PART_STRIDE +
                         (size_t)(kxp * 32 + my0 + j) * 2048 + q * 4;
      unsigned ldsoff = lds_base + (unsigned)(plane * 2048 + q * 4) * 4u;
      asm volatile("global_load_async_to_lds_b128 %0, %1, off"
                   :
                   : "v"(ldsoff), "v"(src)
                   : "memory");
    }
    asm volatile("s_wait_asynccnt 0" ::: "memory");
  }
  __syncthreads();

  const int w = tid >> 5;
  const int l = tid & 31;
  const int ln = l & 15;  // A row m (=batch) and B col n (=o offset)
  const int h = l >> 4;
  const int o0 = w * 16;
  const int mx = kxp & 31;
  const float* wr = (kxp < 32) ? wr1 : wr2;
  const float* wi = (kxp < 32) ? wi1 : wi2;
  const size_t colbase = (size_t)mx * 32 + my0;

  v8f accre[8], accim[8];
#pragma unroll
  for (int j = 0; j < 8; ++j) {
    accre[j] = vzero8();
    accim[j] = vzero8();
  }

  for (int k = 0; k < 128; k += 4) {
    const int r0 = k + 2 * h;  // K rows for this lane's half (A and B agree)
    const int r1 = r0 + 1;
    const size_t off0 = (size_t)(r0 * 128 + o0 + ln) * 1024 + colbase;
    const size_t off1 = (size_t)(r1 * 128 + o0 + ln) * 1024 + colbase;
    float wre0[8], wre1[8], wim0[8], wim1[8];
    *reinterpret_cast<float4*>(&wre0[0]) = *reinterpret_cast<const float4*>(wr + off0);
    *reinterpret_cast<float4*>(&wre0[4]) = *reinterpret_cast<const float4*>(wr + off0 + 4);
    *reinterpret_cast<float4*>(&wre1[0]) = *reinterpret_cast<const float4*>(wr + off1);
    *reinterpret_cast<float4*>(&wre1[4]) = *reinterpret_cast<const float4*>(wr + off1 + 4);
    *reinterpret_cast<float4*>(&wim0[0]) = *reinterpret_cast<const float4*>(wi + off0);
    *reinterpret_cast<float4*>(&wim0[4]) = *reinterpret_cast<const float4*>(wi + off0 + 4);
    *reinterpret_cast<float4*>(&wim1[0]) = *reinterpret_cast<const float4*>(wi + off1);
    *reinterpret_cast<float4*>(&wim1[4]) = *reinterpret_cast<const float4*>(wi + off1 + 4);
    if (k + 4 < 128) {
      __builtin_prefetch(wr + off0 + (size_t)4 * 128 * 1024, 0, 1);
      __builtin_prefetch(wi + off0 + (size_t)4 * 128 * 1024, 0, 1);
    }
#pragma unroll
    for (int j = 0; j < 8; ++j) {
      v2f are, aim, bre, bim, bimn;
      are.x = lds[(2 * j) * 2048 + r0 * 16 + ln];
      are.y = lds[(2 * j) * 2048 + r1 * 16 + ln];
      aim.x = lds[(2 * j + 1) * 2048 + r0 * 16 + ln];
      aim.y = lds[(2 * j + 1) * 2048 + r1 * 16 + ln];
      bre.x = wre0[j];
      bre.y = wre1[j];
      bim.x = wim0[j];
      bim.y = wim1[j];
      bimn = -bim;
      // C_re += Are*Wre - Aim*Wim ; C_im += Are*Wim + Aim*Wre
      accre[j] = wmma4(are, bre, accre[j]);
      accre[j] = wmma4(aim, bimn, accre[j]);
      accim[j] = wmma4(are, bim, accim[j]);
      accim[j] = wmma4(aim, bre, accim[j]);
    }
  }

  if (!FINAL) {
    // vout[part][mode][o][b]; lane holds b = v+8h for its (o = o0+ln)
#pragma unroll
    for (int j = 0; j < 8; ++j) {
      const size_t mode = (size_t)(kxp * 32 + my0 + j);
      float* base_re = vout + mode * 2048 + (size_t)(o0 + ln) * 16 + 8 * h;
      float* base_im = base_re + PART_STRIDE;
      v8f cr = accre[j], ci = accim[j];
      *reinterpret_cast<float4*>(base_re) = make_float4(cr[0], cr[1], cr[2], cr[3]);
      *reinterpret_cast<float4*>(base_re + 4) = make_float4(cr[4], cr[5], cr[6], cr[7]);
      *reinterpret_cast<float4*>(base_im) = make_float4(ci[0], ci[1], ci[2], ci[3]);
      *reinterpret_cast<float4*>(base_im + 4) = make_float4(ci[4], ci[5], ci[6], ci[7]);
    }
  } else {
    // vout[b][part][c][mode] for the fused final kernel
#pragma unroll
    for (int j = 0; j < 8; ++j) {
      const int mode = kxp * 32 + my0 + j;
      v8f cr = accre[j], ci = accim[j];
#pragma unroll
      for (int v = 0; v < 8; ++v) {
        int bb = v + 8 * h;
        size_t idr = (((size_t)bb * 2 + 0) * 128 + (o0 + ln)) * 2048 + mode;
        size_t idi = (((size_t)bb * 2 + 1) * 128 + (o0 + ln)) * 2048 + mode;
        vout[idr] = cr[v];
        vout[idi] = ci[v];
      }
    }
  }
}

// ----- Kernel 4: Hermitian symmetrization of ky=0 column (inter-layer) -----
__global__ __launch_bounds__(256) void fno_sym(float* __restrict__ vf) {
  int idx = blockIdx.x * 256 + threadIdx.x;  // 67584 = 33*128*16
  if (idx >= 67584) return;
  int b = idx & 15;
  int i = (idx >> 4) & 127;
  int g = idx >> 11;  // 0..32
  size_t off = (size_t)i * 16 + b;
  if (g == 0) {
    vf[PART_STRIDE + off] = 0.f;  // Im(kx=0, ky=0) -> 0
  } else if (g == 32) {
    size_t m = (size_t)1024 * 2048 + off;  // kxp=32 (kx=96), mirror is zero
    vf[m] *= 0.5f;
    vf[PART_STRIDE + m] *= 0.5f;
  } else {
    size_t ik = (size_t)(g * 32) * 2048 + off;
    size_t im = (size_t)((64 - g) * 32) * 2048 + off;
    float rk = vf[ik], rm = vf[im];
    float xk = vf[PART_STRIDE + ik], xm = vf[PART_STRIDE + im];
    float nr = 0.5f * (rk + rm);
    float ni = 0.5f * (xk - xm);
    vf[ik] = nr;
    vf[im] = nr;
    vf[PART_STRIDE + ik] = ni;
    vf[PART_STRIDE + im] = -ni;
  }
}

// --- Kernel 5: fused x-iDFT + pw1 GEMM (WMMA) + y-c2r + gelu + pw2 dot ------
// One block per (b, x). of layout: [b][part][c][kxp*32+ky]
__global__ __launch_bounds__(256) void fno_final(const float* __restrict__ of,
                                                 const float* __restrict__ pw1,
                                                 const float* __restrict__ pb1,
                                                 const float* __restrict__ pw2,
                                                 const float* __restrict__ pb2,
                                                 float* __restrict__ out) {
  extern __shared__ float lds[];
  float* ct = lds;                // 128
  float* st = lds + 128;          // 128
  float* g_re = lds + 256;        // 128c x 32ky
  float* g_im = g_re + 4096;
  float* G_re = g_im + 4096;      // 128p x 32ky
  float* G_im = G_re + 4096;
  float* partial = G_im + 4096;   // 256
  const int b = blockIdx.x >> 7;
  const int x = blockIdx.x & 127;
  const int tid = threadIdx.x;

  if (tid < 128) {
    float s, c;
    __sincosf(TWO_PI_OVER_N * (float)tid, &s, &c);
    ct[tid] = c;
    st[tid] = s;
  }
  __syncthreads();

  // x-axis inverse DFT: g[c,ky] = (1/128) * sum_kx of[c,kx,ky] * e^{+i*2pi*kx*x/128}
  {
    const float* ofb = of + (size_t)b * 524288;
    int ky = tid & 31;
    int cg = tid >> 5;
    for (int jj = 0; jj < 16; ++jj) {
      int c = cg * 16 + jj;
      const float* pre = ofb + (size_t)c * 2048;
      const float* pim = ofb + 262144 + (size_t)c * 2048;
      float sre = 0.f, sim = 0.f;
      for (int kxp = 0; kxp < 64; ++kxp) {
        float re = pre[kxp * 32 + ky];
        float im = pim[kxp * 32 + ky];
        int kx = kxp + ((kxp >= 32) ? 64 : 0);
        int ph = (kx * x) & 127;
        float cc = ct[ph], ss = st[ph];
        sre += re * cc - im * ss;
        sim += re * ss + im * cc;
      }
      g_re[c * 32 + ky] = sre * (1.0f / 128.0f);
      g_im[c * 32 + ky] = sim * (1.0f / 128.0f);
    }
  }
  __syncthreads();

  // G[p,col] = sum_c pw1[p,c] * g[c,col], cols 0..31 = re, 32..63 = im (WMMA)
  {
    const int w = tid >> 5;
    const int l = tid & 31;
    const int m = l & 15;
    const int h = l >> 4;
    const int p = w * 16 + m;
    v8f acc[4];
#pragma unroll
    for (int t = 0; t < 4; ++t) acc[t] = vzero8();
    for (int k = 0; k < 128; k += 4) {
      int r0 = k + 2 * h;
      int r1 = r0 + 1;
      v2f a;
      a.x = pw1[p * 128 + r0];
      a.y = pw1[p * 128 + r1];
#pragma unroll
      for (int t = 0; t < 4; ++t) {
        int col = t * 16 + m;
        v2f bb;
        bb.x = (col < 32) ? g_re[r0 * 32 + col] : g_im[r0 * 32 + col - 32];
        bb.y = (col < 32) ? g_re[r1 * 32 + col] : g_im[r1 * 32 + col - 32];
        acc[t] = wmma4(a, bb, acc[t]);
      }
    }
#pragma unroll
    for (int t = 0; t < 4; ++t) {
#pragma unroll
      for (int v = 0; v < 8; ++v) {
        int col = t * 16 + m;
        int p2 = w * 16 + v + 8 * h;
        if (col < 32)
          G_re[p2 * 32 + col] = acc[t][v];
        else
          G_im[p2 * 32 + col - 32] = acc[t][v];
      }
    }
  }
  __syncthreads();

  // y-axis c2r + bias + gelu + pw2 reduction
  {
    int y = tid & 127;
    int half = tid >> 7;
    float s = 0.f;
    for (int pp = half * 64; pp < half * 64 + 64; ++pp) {
      float hr = G_re[pp * 32 + 0];  // DC: imag ignored (c2r semantics)
      for (int k = 1; k < 32; ++k) {
        int ph = (k * y) & 127;
        hr += 2.0f * (G_re[pp * 32 + k] * ct[ph] - G_im[pp * 32 + k] * st[ph]);
      }
      hr = hr * (1.0f / 128.0f) + pb1[pp];
      float u = 0.7978845608028654f * (hr + 0.044715f * hr * hr * hr);
      float gel = 0.5f * hr * (1.0f + tanhf(u));
      s += gel * pw2[pp];
    }
    partial[tid] = s;
  }
  __syncthreads();
  if (tid < 128)
    out[(size_t)b * 16384 + x * 128 + tid] =
        partial[tid] + partial[128 + tid] + pb2[0];
}

// ---------------------------------------------------------------------------
extern "C" void kernel_launch(void* const* d_in, const int* in_sizes, int n_in,
                              void* d_out, int out_size, void* d_ws,
                              size_t ws_size, hipStream_t stream) {
  (void)in_sizes; (void)n_in; (void)out_size; (void)ws_size;
  const float* z = (const float*)d_in[0];
  const float* lift_w = (const float*)d_in[1];
  const float* lift_b = (const float*)d_in[2];
  const float* wr1 = (const float*)d_in[3];
  const float* wi1 = (const float*)d_in[4];
  const float* wr2 = (const float*)d_in[5];
  const float* wi2 = (const float*)d_in[6];
  const float* pw1 = (const float*)d_in[7];
  const float* pb1 = (const float*)d_in[8];
  const float* pw2 = (const float*)d_in[9];
  const float* pb2 = (const float*)d_in[10];

  float* ws = (float*)d_ws;
  float* vfA = ws;                    // 8388608 floats
  float* vfB = ws + 8388608;          // 8388608 floats
  float* Rre = ws + 16777216;         // 65536
  float* Rim = Rre + 65536;           // 65536
  float* Zre = Rim + 65536;           // 32768
  float* Zim = Zre + 32768;           // 32768

  hipFuncSetAttribute((const void*)fno_mix<false>,
                      hipFuncAttributeMaxDynamicSharedMemorySize, 131072);
  hipFuncSetAttribute((const void*)fno_mix<true>,
                      hipFuncAttributeMaxDynamicSharedMemorySize, 131072);
  hipFuncSetAttribute((const void*)fno_final,
                      hipFuncAttributeMaxDynamicSharedMemorySize, 67584);

  fno_dft_rows<<<256, 256, 0, stream>>>(z, Rre, Rim);
  fno_dft_cols<<<128, 256, 0, stream>>>(Rre, Rim, Zre, Zim);
  fno_lift<<<32768, 256, 0, stream>>>(Zre, Zim, lift_w, lift_b, vfA);

  const size_t WL = 16777216;  // per-layer weight elements (128*128*32*32)
  for (int l = 0; l < 4; ++l) {
    const float* vin = (l & 1) ? vfB : vfA;
    float* vout = (l & 1) ? vfA : vfB;
    if (l < 3) {
      fno_mix<false><<<256, 256, 131072, stream>>>(
          vin, vout, wr1 + l * WL, wi1 + l * WL, wr2 + l * WL, wi2 + l * WL);
      fno_sym<<<264, 256, 0, stream>>>(vout);
    } else {
      fno_mix<true><<<256, 256, 131072, stream>>>(
          vin, vout, wr1 + l * WL, wi1 + l * WL, wr2 + l * WL, wi2 + l * WL);
    }
  }
  // 4 layers: A->B->A->B->A ; final spectrum (final layout) lives in vfA
  fno_final<<<2048, 256, 67584, stream>>>(vfA, pw1, pb1, pw2, pb2,
                                          (float*)d_out);
}